// S4Layer_18047452578305
// MI455X (gfx1250) — compile-verified
//
#include <hip/hip_runtime.h>
#include <hip/hip_bf16.h>

// S4 layer forward for MI455X (gfx1250).
// - FFT-domain convolution (the ~1 GFLOP algorithm; a direct Toeplitz matmul
//   would be ~4 GFLOP causal). Entire footprint is L2-resident (192 MB).
// - Cauchy reduction done as complex [16x64]x[64x4] GEMMs on the fp32 matrix
//   pipe: V_WMMA_F32_16X16X4_F32 accumulation chains.
// - DIF forward -> pointwise (bit-reversed order) -> DIT inverse: no
//   bit-reversal passes anywhere.
// - Global traffic made coalesced via tiled LDS transposes (u is [B,L,H] with
//   a 1 KB h-stride; FFTs want contiguous L rows).

typedef __attribute__((ext_vector_type(2))) float v2f;
typedef __attribute__((ext_vector_type(8))) float v8f;

namespace {
constexpr int kH = 256;
constexpr int kN = 64;
constexpr int kL = 2048;
constexpr int kNF = 4096;  // padded FFT length
constexpr int kB = 4;
constexpr float kPI = 3.14159265358979323846f;
}

// rcp + one Newton-Raphson step: ~fp32-accurate, avoids v_div_* sequences.
__device__ __forceinline__ float fast_rcp(float x) {
  float r = __builtin_amdgcn_rcpf(x);
  float e = __builtin_fmaf(-x, r, 1.0f);
  return __builtin_fmaf(r, e, r);
}

// ---------------------------------------------------------------------------
// In-place radix-2 DIF FFT (forward, e^{-i}), natural in -> bit-reversed out.
// tw[j] = e^{-2 pi i j / kNF}, j in [0, kNF/2). Valid for any n dividing kNF.
// Caller must __syncthreads() before entry.
// ---------------------------------------------------------------------------
__device__ __forceinline__ void fft_dif(float2* X, const float2* tw, int n,
                                        int tid, int nthreads) {
  const int nb = n >> 1;
  int hbits = 31 - __clz(nb);
  int step = kNF / n;  // constant at call sites
  for (int len = n; len >= 2; len >>= 1, --hbits, step <<= 1) {
    const int half = len >> 1;
    for (int idx = tid; idx < nb; idx += nthreads) {
      const int j = idx & (half - 1);
      const int p = ((idx >> hbits) << (hbits + 1)) + j;
      const float2 u = X[p];
      const float2 v = X[p + half];
      const float2 w = tw[j * step];
      const float sr = u.x - v.x, si = u.y - v.y;
      X[p] = make_float2(u.x + v.x, u.y + v.y);
      X[p + half] = make_float2(sr * w.x - si * w.y, sr * w.y + si * w.x);
    }
    __syncthreads();
  }
}

// ---------------------------------------------------------------------------
// In-place radix-2 DIT inverse FFT (e^{+i}), bit-reversed in -> natural out.
// No 1/n scaling (caller applies). Caller syncs before entry.
// ---------------------------------------------------------------------------
__device__ __forceinline__ void fft_dit_inv(float2* X, const float2* tw, int n,
                                            int tid, int nthreads) {
  const int nb = n >> 1;
  int hbits = 0;
  int step = kNF >> 1;
  for (int len = 2; len <= n; len <<= 1, ++hbits, step >>= 1) {
    const int half = len >> 1;
    for (int idx = tid; idx < nb; idx += nthreads) {
      const int j = idx & (half - 1);
      const int p = ((idx >> hbits) << (hbits + 1)) + j;
      const float2 w0 = tw[j * step];
      const float wr = w0.x, wi = -w0.y;  // conjugate -> inverse
      const float2 v0 = X[p + half];
      const float vr = v0.x * wr - v0.y * wi;
      const float vi = v0.x * wi + v0.y * wr;
      const float2 u = X[p];
      X[p] = make_float2(u.x + vr, u.y + vi);
      X[p + half] = make_float2(u.x - vr, u.y - vi);
    }
    __syncthreads();
  }
}

__device__ __forceinline__ void build_twiddles(float2* tw, int tid,
                                               int nthreads) {
  for (int j = tid; j < kNF / 2; j += nthreads) {
    float s, c;
    __sincosf(-2.0f * kPI * (float)j / (float)kNF, &s, &c);
    tw[j] = make_float2(c, s);
  }
}

// ---------------------------------------------------------------------------
// Kernel A: one workgroup per h. Kd[h][k] (complex, bit-reversed over kNF)
//   = FFT_4096( zero-pad( Re(FFT_2048(at_roots[h]))/2048 ) ).
// (roll(flip(ifft(A)),1).real == Re(forwardFFT(A))/L identity.)
// Cauchy stage: per 16-row L tile, two V_WMMA_F32_16X16X4_F32 accumulation
// chains; C cols 0-3 = Re(k00..k11), cols 4-7 = Im(k00..k11).
// ---------------------------------------------------------------------------
__launch_bounds__(256)
__global__ void s4_make_kernel_fft(const float* __restrict__ p_ri,
                                   const float* __restrict__ q_ri,
                                   const float* __restrict__ lam_ri,
                                   const float* __restrict__ B_ri,
                                   const float* __restrict__ Ct_ri,
                                   const float* __restrict__ log_step,
                                   float2* __restrict__ Kd) {
  __shared__ float2 X[kNF];        // 32 KB
  __shared__ float2 tw[kNF / 2];   // 16 KB
  __shared__ float wre[4][kN];
  __shared__ float wim[4][kN];
  __shared__ float lre[kN];
  __shared__ float lim[kN];
  __shared__ float scratch[8][16][8];  // per-wave C-matrix dump

  const int h = blockIdx.x;
  const int tid = threadIdx.x;
  const int nthreads = blockDim.x;  // 256 = 8 waves
  const int lane = tid & 31;
  const int wave = tid >> 5;

  build_twiddles(tw, tid, nthreads);
  const float stp = __expf(log_step[h]);
  if (tid < kN) {
    const int n = tid;
    const float pr = p_ri[2 * n], pi = p_ri[2 * n + 1];
    const float qr = q_ri[2 * n], qi = q_ri[2 * n + 1];
    const float br = B_ri[(h * kN + n) * 2], bi = B_ri[(h * kN + n) * 2 + 1];
    const float cr = Ct_ri[(h * kN + n) * 2], ci = Ct_ri[(h * kN + n) * 2 + 1];
    // w0 = conj(Ct)*B, w1 = conj(Ct)*p, w2 = conj(q)*B, w3 = conj(q)*p
    wre[0][n] = cr * br + ci * bi;  wim[0][n] = cr * bi - ci * br;
    wre[1][n] = cr * pr + ci * pi;  wim[1][n] = cr * pi - ci * pr;
    wre[2][n] = qr * br + qi * bi;  wim[2][n] = qr * bi - qi * br;
    wre[3][n] = qr * pr + qi * pi;  wim[3][n] = qr * pi - qi * pr;
    lre[n] = lam_ri[2 * n];
    lim[n] = lam_ri[2 * n + 1];
  }
  __syncthreads();

  // ---- Cauchy stage via WMMA ----
  const float two_over_step = 2.0f * fast_rcp(stp);
  const int m = lane & 15;        // A row / B,C column
  const int halfsel = lane >> 4;  // K-pair select (ISA 16x4 A/B layout)
  const int jj = m & 3;
  const bool lower8 = (m < 8);
  const bool repart = (m < 4);

  for (int t = wave; t < kL / 16; t += 8) {  // 16 uniform iterations per wave
    const int l = t * 16 + m;
    float sn, cs;
    __sincosf(2.0f * kPI * (float)l / (float)kL, &sn, &cs);
    // g = (2/step) * (1-w)/(1+w)
    const float nr = 1.0f - cs, ni = -sn;
    const float dr = 1.0f + cs, di = sn;
    const float invd = fast_rcp(dr * dr + di * di);
    const float gre = two_over_step * (nr * dr + ni * di) * invd;
    const float gim = two_over_step * (ni * dr - nr * di) * invd;

    v8f acc = {0.f, 0.f, 0.f, 0.f, 0.f, 0.f, 0.f, 0.f};
    for (int ks = 0; ks < 16; ++ks) {
      const int n0 = ks * 4 + 2 * halfsel;  // this lane's two K indices
      // A operands: r = 1/(g - lambda) for n0, n0+1
      v2f a_re, a_im;
      {
        const float d0r = gre - lre[n0], d0i = gim - lim[n0];
        const float iv0 = fast_rcp(d0r * d0r + d0i * d0i);
        const float d1r = gre - lre[n0 + 1], d1i = gim - lim[n0 + 1];
        const float iv1 = fast_rcp(d1r * d1r + d1i * d1i);
        a_re.x = d0r * iv0;  a_im.x = -d0i * iv0;
        a_re.y = d1r * iv1;  a_im.y = -d1i * iv1;
      }
      // B operands (unconditional loads + selects; no EXEC branching):
      // cols 0-3 <- W_re / -W_im ; cols 4-7 <- W_im / W_re ; cols 8-15 zero.
      const float wr0 = wre[jj][n0], wi0 = wim[jj][n0];
      const float wr1 = wre[jj][n0 + 1], wi1 = wim[jj][n0 + 1];
      v2f b1, b2;
      b1.x = lower8 ? (repart ? wr0 : wi0) : 0.0f;
      b1.y = lower8 ? (repart ? wr1 : wi1) : 0.0f;
      b2.x = lower8 ? (repart ? -wi0 : wr0) : 0.0f;
      b2.y = lower8 ? (repart ? -wi1 : wr1) : 0.0f;

      acc = __builtin_amdgcn_wmma_f32_16x16x4_f32(false, a_re, false, b1,
                                                  (short)0, acc, false, false);
      acc = __builtin_amdgcn_wmma_f32_16x16x4_f32(false, a_im, false, b2,
                                                  (short)0, acc, false, false);
    }
    // C layout: VGPR r -> rows r (lanes 0-15) and r+8 (lanes 16-31)
#pragma unroll
    for (int r = 0; r < 8; ++r) {
      if (lower8) scratch[wave][r + 8 * halfsel][m] = acc[r];
    }
    __syncthreads();
    if (lane < 16) {
      const float k00r = scratch[wave][lane][0];
      const float k01r = scratch[wave][lane][1];
      const float k10r = scratch[wave][lane][2];
      const float k11r = scratch[wave][lane][3];
      const float k00i = scratch[wave][lane][4];
      const float k01i = scratch[wave][lane][5];
      const float k10i = scratch[wave][lane][6];
      const float k11i = scratch[wave][lane][7];
      // c = 2/(1+w); reuse this lane's sincos (l == t*16+lane for lanes 0-15)
      const float cdr = 1.0f + cs, cdi = sn;
      const float civ = 2.0f * fast_rcp(cdr * cdr + cdi * cdi);
      const float cre = cdr * civ, cie = -cdi * civ;
      // at = c * (k00 - k01*k10/(1+k11))
      const float er = 1.0f + k11r, ei = k11i;
      const float eiv = fast_rcp(er * er + ei * ei);
      const float nr2 = k01r * k10r - k01i * k10i;
      const float ni2 = k01r * k10i + k01i * k10r;
      const float fr = (nr2 * er + ni2 * ei) * eiv;
      const float fi = (ni2 * er - nr2 * ei) * eiv;
      const float vr = k00r - fr, vi = k00i - fi;
      X[l] = make_float2(cre * vr - cie * vi, cre * vi + cie * vr);
    }
    __syncthreads();
  }
  __syncthreads();

  // ---- K[l] = Re(FFT_2048(at_roots)[l]) / 2048 ----
  fft_dif(X, tw, kL, tid, nthreads);  // bit-reversed out
  float keep[8];
#pragma unroll
  for (int i = 0; i < 8; ++i) {
    const int l = tid + i * nthreads;
    keep[i] = X[__brev((unsigned)l) >> 21].x;  // 11-bit reverse read
  }
  __syncthreads();
  const float sc = 1.0f / (float)kL;
#pragma unroll
  for (int i = 0; i < 8; ++i) {
    const int l = tid + i * nthreads;
    X[l] = make_float2(keep[i] * sc, 0.0f);
    X[l + kL] = make_float2(0.0f, 0.0f);
  }
  __syncthreads();

  // ---- Kd = FFT_4096(K zero-padded), kept bit-reversed ----
  fft_dif(X, tw, kNF, tid, nthreads);
  for (int k = tid; k < kNF; k += nthreads) {
    Kd[(size_t)h * kNF + k] = X[k];  // coalesced b64 stores
  }
}

// ---------------------------------------------------------------------------
// Convolution kernel: one workgroup per (h, b).
// transposed==1: uin/yout are [B][H][L] (contiguous rows, coalesced);
//                D*u epilogue is fused into the transpose-back kernel.
// transposed==0: uin/yout are [B][L][H] (strided fallback), D*u fused here.
// ---------------------------------------------------------------------------
__launch_bounds__(256)
__global__ void s4_conv_fft(const float* __restrict__ uin,
                            const float2* __restrict__ Kd,
                            const float* __restrict__ D,
                            float* __restrict__ yout, int transposed) {
  __shared__ float2 X[kNF];       // 32 KB
  __shared__ float2 tw[kNF / 2];  // 16 KB

  const int h = blockIdx.x;
  const int b = blockIdx.y;
  const int tid = threadIdx.x;
  const int nthreads = blockDim.x;

  const size_t base = transposed ? ((size_t)b * kH + h) * kL
                                 : ((size_t)b * kL) * kH + h;
  const int stride = transposed ? 1 : kH;

  build_twiddles(tw, tid, nthreads);
  for (int l = tid; l < kL; l += nthreads) {
    X[l] = make_float2(uin[base + (size_t)l * stride], 0.0f);
    X[l + kL] = make_float2(0.0f, 0.0f);
  }
  __syncthreads();

  fft_dif(X, tw, kNF, tid, nthreads);

  for (int k = tid; k < kNF; k += nthreads) {
    const float2 kd = Kd[(size_t)h * kNF + k];  // same bit-reversed order
    const float2 x = X[k];
    X[k] = make_float2(x.x * kd.x - x.y * kd.y, x.x * kd.y + x.y * kd.x);
  }
  __syncthreads();

  fft_dit_inv(X, tw, kNF, tid, nthreads);

  const float isc = 1.0f / (float)kNF;
  if (transposed) {
    for (int l = tid; l < kL; l += nthreads) {
      yout[base + l] = X[l].x * isc;
    }
  } else {
    const float dv = D[h];
    for (int l = tid; l < kL; l += nthreads) {
      const size_t idx = base + (size_t)l * stride;
      yout[idx] = X[l].x * isc + dv * uin[idx];
    }
  }
}

// ---------------------------------------------------------------------------
// Tiled transposes (32x33 LDS tile, coalesced both directions).
// ---------------------------------------------------------------------------
__launch_bounds__(256)
__global__ void s4_transpose_in(const float* __restrict__ u,
                                float* __restrict__ ut) {
  // u: [B][L][H] -> ut: [B][H][L]; grid (H/32, L/32, B), block (32, 8)
  __shared__ float tile[32][33];
  const int b = blockIdx.z;
  const int h0 = blockIdx.x * 32;
  const int l0 = blockIdx.y * 32;
  for (int r = threadIdx.y; r < 32; r += 8) {
    tile[r][threadIdx.x] =
        u[((size_t)b * kL + (l0 + r)) * kH + (h0 + threadIdx.x)];
  }
  __syncthreads();
  for (int r = threadIdx.y; r < 32; r += 8) {
    ut[((size_t)b * kH + (h0 + r)) * kL + (l0 + threadIdx.x)] =
        tile[threadIdx.x][r];
  }
}

__launch_bounds__(256)
__global__ void s4_transpose_out(const float* __restrict__ yt,
                                 const float* __restrict__ u,
                                 const float* __restrict__ D,
                                 float* __restrict__ out) {
  // yt: [B][H][L] -> out: [B][L][H] with out += D[h]*u; grid (L/32, H/32, B)
  __shared__ float tile[32][33];
  const int b = blockIdx.z;
  const int l0 = blockIdx.x * 32;
  const int h0 = blockIdx.y * 32;
  for (int r = threadIdx.y; r < 32; r += 8) {
    tile[r][threadIdx.x] =
        yt[((size_t)b * kH + (h0 + r)) * kL + (l0 + threadIdx.x)];
  }
  __syncthreads();
  for (int r = threadIdx.y; r < 32; r += 8) {
    const int h = h0 + threadIdx.x;
    const size_t idx = ((size_t)b * kL + (l0 + r)) * kH + h;
    out[idx] = tile[threadIdx.x][r] + D[h] * u[idx];
  }
}

// ---------------------------------------------------------------------------
extern "C" void kernel_launch(void* const* d_in, const int* in_sizes, int n_in,
                              void* d_out, int out_size, void* d_ws,
                              size_t ws_size, hipStream_t stream) {
  const float* u = (const float*)d_in[0];
  const float* p_ri = (const float*)d_in[1];
  const float* q_ri = (const float*)d_in[2];
  const float* lam_ri = (const float*)d_in[3];
  const float* B_ri = (const float*)d_in[4];
  const float* Ct_ri = (const float*)d_in[5];
  const float* D = (const float*)d_in[6];
  const float* log_step = (const float*)d_in[7];
  float* out = (float*)d_out;

  const size_t kdBytes = (size_t)kH * kNF * sizeof(float2);       // 8 MB
  const size_t matBytes = (size_t)kB * kH * kL * sizeof(float);   // 8 MB
  float2* Kd = (float2*)d_ws;

  s4_make_kernel_fft<<<dim3(kH), dim3(256), 0, stream>>>(
      p_ri, q_ri, lam_ri, B_ri, Ct_ri, log_step, Kd);

  if (ws_size >= kdBytes + 2 * matBytes) {
    // Coalesced path: transpose -> contiguous-row FFT conv -> transpose+D*u.
    float* ut = (float*)((char*)d_ws + kdBytes);
    float* yt = (float*)((char*)d_ws + kdBytes + matBytes);
    s4_transpose_in<<<dim3(kH / 32, kL / 32, kB), dim3(32, 8), 0, stream>>>(
        u, ut);
    s4_conv_fft<<<dim3(kH, kB), dim3(256), 0, stream>>>(ut, Kd, D, yt, 1);
    s4_transpose_out<<<dim3(kL / 32, kH / 32, kB), dim3(32, 8), 0, stream>>>(
        yt, u, D, out);
  } else {
    // Strided fallback (still correct, worse vmem request count).
    s4_conv_fft<<<dim3(kH, kB), dim3(256), 0, stream>>>(u, Kd, D, out, 0);
  }
}